// TopKPooling_54425825575647
// MI455X (gfx1250) — compile-verified
//
#include <hip/hip_runtime.h>
#include <math.h>

// ---------------- problem constants (match reference) ----------------
#define BG   128                  // graphs
#define NN   512                  // nodes per graph
#define CC   256                  // channels
#define DEG  32
#define EE   (BG * NN * DEG)      // 2,097,152 edges
#define KK   256                  // kept per graph
#define NTOT (BG * NN)            // 65,536 nodes
#define BK   (BG * KK)            // 32,768 kept nodes

typedef float v2f __attribute__((ext_vector_type(2)));
typedef float v8f __attribute__((ext_vector_type(8)));

// ---------------- kernel 0: 1/||w|| -------------------------------------
__global__ void wnorm_kernel(const float* __restrict__ w, float* __restrict__ rinv) {
    int lane = threadIdx.x & 31;
    float ss = 0.f;
    #pragma unroll
    for (int j = 0; j < CC / 32; ++j) {
        float v = w[lane + 32 * j];
        ss += v * v;
    }
    #pragma unroll
    for (int off = 16; off > 0; off >>= 1)
        ss += __shfl_xor(ss, off, 32);
    if (lane == 0) rinv[0] = 1.0f / sqrtf(ss);
}

// ---------------- kernel 1: score = tanh((X.w)/||w||) via WMMA f32 -------
// One wave per 16-row tile. A = 16x4 fp32 slab of X; B = 4x16 with the
// weight vector in column N=0 only (lanes 0 and 16 hold B per ISA layout).
// After 64 accumulation steps, column 0 of the 16x16 D matrix lives in
// lanes 0 (M=0..7) and 16 (M=8..15), VGPRs 0..7.
__global__ void score_kernel(const float* __restrict__ x,
                             const float* __restrict__ w,
                             const float* __restrict__ rinv_p,
                             float* __restrict__ score) {
    const int lane   = threadIdx.x & 31;
    const int waveId = threadIdx.x >> 5;
    const int tile   = blockIdx.x * (blockDim.x >> 5) + waveId;
    const int rowBase = tile * 16;

    const int m    = lane & 15;        // matrix row within tile
    const int koff = (lane >> 4) << 1; // 0 for lanes 0-15, 2 for lanes 16-31

    const float rinv = rinv_p[0];
    const float* xrow = x + (size_t)(rowBase + m) * CC + koff;

    v8f c = {0.f, 0.f, 0.f, 0.f, 0.f, 0.f, 0.f, 0.f};

    #pragma unroll 4
    for (int k0 = 0; k0 < CC; k0 += 4) {
        const float2 av = *(const float2*)(xrow + k0);
        const float2 wv = *(const float2*)(w + k0 + koff);
        v2f a; a.x = av.x;                 a.y = av.y;
        v2f b; b.x = (m == 0) ? wv.x : 0.f; b.y = (m == 0) ? wv.y : 0.f;
        // D = A(16x4) * B(4x16) + C ; emits v_wmma_f32_16x16x4_f32
        c = __builtin_amdgcn_wmma_f32_16x16x4_f32(
                false, a, false, b, (short)0, c, false, false);
    }

    if (m == 0) {                          // lanes 0 and 16 hold column N=0
        const int mrow = (lane == 0) ? 0 : 8;
        #pragma unroll
        for (int r = 0; r < 8; ++r)
            score[rowBase + mrow + r] = tanhf(c[r] * rinv);
    }
}

// ---------------- kernel 2: per-graph bitonic top-K ----------------------
// One block (512 threads) per graph. Stable descending order is emulated by
// breaking score ties with the smaller original index first.
__global__ void topk_kernel(const float* __restrict__ score,
                            int*  __restrict__ perm_ws,
                            float* __restrict__ out_perm,
                            float* __restrict__ out_batch) {
    __shared__ float s_key[NN];
    __shared__ int   s_idx[NN];
    const int t = threadIdx.x;
    const int g = blockIdx.x;

    s_key[t] = score[g * NN + t];
    s_idx[t] = t;
    __syncthreads();

    for (int k = 2; k <= NN; k <<= 1) {
        for (int j = k >> 1; j > 0; j >>= 1) {
            const int ixj = t ^ j;
            if (ixj > t) {
                const float ka = s_key[t], kb = s_key[ixj];
                const int   ia = s_idx[t], ib = s_idx[ixj];
                // true if element at t should precede element at ixj
                // in the final (descending, stable) order
                const bool pred = (ka > kb) || (ka == kb && ia < ib);
                const bool doSwap = ((t & k) == 0) ? (!pred) : pred;
                if (doSwap) {
                    s_key[t] = kb; s_key[ixj] = ka;
                    s_idx[t] = ib; s_idx[ixj] = ia;
                }
            }
            __syncthreads();
        }
    }

    if (t < KK) {
        const int p = g * NN + s_idx[t];
        perm_ws [g * KK + t] = p;
        out_perm [g * KK + t] = (float)p;
        out_batch[g * KK + t] = (float)g;
    }
}

// ---------------- kernel 3: gather rows, scale by score ------------------
// One wave per kept row; float4 (b128) traffic. X is L2-resident (64MB<192MB).
__global__ void gather_kernel(const float* __restrict__ x,
                              const float* __restrict__ score,
                              const int*   __restrict__ perm_ws,
                              float* __restrict__ x_out,
                              float* __restrict__ x_ae) {
    const int lane = threadIdx.x & 31;
    const int wid  = threadIdx.x >> 5;
    const int row  = blockIdx.x * (blockDim.x >> 5) + wid;
    if (row >= BK) return;

    const int   src = perm_ws[row];
    const float sc  = score[src];

    const float4* xin = (const float4*)(x + (size_t)src * CC);
    float4* oS = (float4*)(x_out + (size_t)row * CC);
    float4* oR = (float4*)(x_ae  + (size_t)row * CC);

    #pragma unroll
    for (int j = lane; j < CC / 4; j += 32) {
        float4 v = xin[j];
        oR[j] = v;
        float4 vs; vs.x = v.x * sc; vs.y = v.y * sc; vs.z = v.z * sc; vs.w = v.w * sc;
        oS[j] = vs;
    }
}

// ---------------- kernel 4a/4b: node_map build ---------------------------
__global__ void fill_kernel(int* __restrict__ node_map) {
    const int i = blockIdx.x * blockDim.x + threadIdx.x;
    if (i < NTOT) node_map[i] = -1;
}
__global__ void scatter_kernel(const int* __restrict__ perm_ws,
                               int* __restrict__ node_map) {
    const int i = blockIdx.x * blockDim.x + threadIdx.x;
    if (i < BK) node_map[perm_ws[i]] = i;
}

// ---------------- kernel 5: edge remap + mask ----------------------------
__global__ void edge_kernel(const int* __restrict__ ei,       // [2*E] int32
                            const int* __restrict__ node_map,
                            float* __restrict__ out_ei,       // [2*E]
                            float* __restrict__ out_mask) {   // [E]
    const int e = blockIdx.x * blockDim.x + threadIdx.x;
    if (e >= EE) return;
    const int r = node_map[ei[e]];
    const int c = node_map[ei[EE + e]];
    const bool keep = (r >= 0) && (c >= 0);
    out_ei[e]       = (float)(keep ? r : -1);
    out_ei[EE + e]  = (float)(keep ? c : -1);
    out_mask[e]     = keep ? 1.0f : 0.0f;
}

// ---------------- launcher ----------------------------------------------
extern "C" void kernel_launch(void* const* d_in, const int* in_sizes, int n_in,
                              void* d_out, int out_size, void* d_ws, size_t ws_size,
                              hipStream_t stream) {
    const float* x      = (const float*)d_in[0];   // [NTOT, C]
    const int*   ei     = (const int*)  d_in[1];   // [2, E]
    // d_in[2] = batch (unused; batch_out derived from graph id)
    const float* weight = (const float*)d_in[3];   // [1, C]

    // workspace layout
    float* score_ws = (float*)d_ws;                    // NTOT floats
    int*   perm_ws  = (int*)(score_ws + NTOT);         // BK ints
    int*   node_map = perm_ws + BK;                    // NTOT ints
    float* rinv_ws  = (float*)(node_map + NTOT);       // 1 float

    // output layout (floats, flat, reference tuple order)
    float* o = (float*)d_out;
    float* out_x    = o;                                   // BK*C
    float* out_ei   = out_x   + (size_t)BK * CC;           // 2*E
    float* out_mask = out_ei  + (size_t)2 * EE;            // E
    float* out_bat  = out_mask + (size_t)EE;               // BK
    float* out_perm = out_bat + (size_t)BK;                // BK
    float* out_xae  = out_perm + (size_t)BK;               // BK*C

    wnorm_kernel<<<1, 32, 0, stream>>>(weight, rinv_ws);

    // 16 rows per wave, 8 waves per block -> 128 rows/block
    score_kernel<<<NTOT / 128, 256, 0, stream>>>(x, weight, rinv_ws, score_ws);

    topk_kernel<<<BG, NN, 0, stream>>>(score_ws, perm_ws, out_perm, out_bat);

    gather_kernel<<<BK / 8, 256, 0, stream>>>(x, score_ws, perm_ws, out_x, out_xae);

    fill_kernel<<<NTOT / 256, 256, 0, stream>>>(node_map);
    scatter_kernel<<<BK / 256, 256, 0, stream>>>(perm_ws, node_map);

    edge_kernel<<<EE / 256, 256, 0, stream>>>(ei, node_map, out_ei, out_mask);
}